// NGPGridSampler_27144193310746
// MI455X (gfx1250) — compile-verified
//
#include <hip/hip_runtime.h>
#include <cstdint>
#include <cstddef>

#define NGRID     128
#define NCASC     3
#define G_CELLS   (NCASC * NGRID * NGRID * NGRID)   // 6,291,456
#define G_QUADS   (G_CELLS / 4)                     // 1,572,864
#define NRAYS     4096
#define MAXSTEP   1024
#define CONE_ANG  0.00390625f
#define DT_MIN_F  (1.7320508f / 1024.0f)
#define DT_MAX_F  (DT_MIN_F * 16.0f)
#define NEAR_F    0.2f

typedef __attribute__((ext_vector_type(2))) float v2f;
typedef __attribute__((ext_vector_type(8))) float v8f;

// ---------------------------------------------------------------------------
// Wave-level sum of 32 lane values.
// WMMA path: A = ones(16x4) -> D[m,n] = sum_k B[k,n] (column sums). Each lane
// contributes its value to exactly one distinct B slot (b0=s, b1=0); summing
// the 16 column sums via shfl_xor(1,2,4,8) in each 16-lane half yields the
// exact 32-lane total, independent of B's lane->(k,n) permutation. Requires
// EXEC all-ones (full block, straight-line code).
// ---------------------------------------------------------------------------
__device__ __forceinline__ float wave_sum32(float s) {
#if defined(__gfx1250__) && __has_builtin(__builtin_amdgcn_wmma_f32_16x16x4_f32)
    v2f a = {1.0f, 1.0f};
    v2f b = {s, 0.0f};
    v8f c = {};
    c = __builtin_amdgcn_wmma_f32_16x16x4_f32(false, a, false, b, (short)0, c,
                                              false, false);
    float v = c[0];                     // column sum for this lane's column
#else
    float v = s;
    v += __shfl_xor(v, 16, 32);
#endif
    v += __shfl_xor(v, 1, 32);
    v += __shfl_xor(v, 2, 32);
    v += __shfl_xor(v, 4, 32);
    v += __shfl_xor(v, 8, 32);
    return v;                           // total of all 32 lanes, in every lane
}

// ---------------------------------------------------------------------------
// 1) grid = bits(0.95 * density_grid)  (all >= 0 -> uint order == float order)
//    vectorized: one uint4 per thread
// ---------------------------------------------------------------------------
__global__ void k_init_grid(const float4* __restrict__ dg4, uint4* __restrict__ gu4) {
    int i = blockIdx.x * 256 + threadIdx.x;
    float4 v = dg4[i];
    uint4 o;
    o.x = __float_as_uint(v.x * 0.95f);
    o.y = __float_as_uint(v.y * 0.95f);
    o.z = __float_as_uint(v.z * 0.95f);
    o.w = __float_as_uint(v.w * 0.95f);
    gu4[i] = o;
}

// ---------------------------------------------------------------------------
// 2) scatter-max of samples (deterministic integer atomic max), b128 loads
// ---------------------------------------------------------------------------
__global__ void k_scatter(const float4* __restrict__ smp4, const int4* __restrict__ idx4,
                          unsigned* __restrict__ gu) {
    int i = blockIdx.x * 256 + threadIdx.x;
    int4   id = idx4[i];
    float4 v  = smp4[i];
    atomicMax(&gu[id.x], __float_as_uint(v.x));
    atomicMax(&gu[id.y], __float_as_uint(v.y));
    atomicMax(&gu[id.z], __float_as_uint(v.z));
    atomicMax(&gu[id.w], __float_as_uint(v.w));
}

// ---------------------------------------------------------------------------
// 3) partial sums for mean (fixed shape -> deterministic); uint4 stream +
//    prefetch; cross-lane reduce partially on the WMMA pipe
// ---------------------------------------------------------------------------
__global__ void k_partial_sum(const uint4* __restrict__ gu4, float* __restrict__ partials) {
    const int tid = threadIdx.x;
    const int gid = blockIdx.x * 256 + tid;
    float s = 0.0f;
    for (int i = gid; i < G_QUADS; i += 1024 * 256) {
        if (i + 1024 * 256 < G_QUADS)
            __builtin_prefetch(&gu4[i + 1024 * 256], 0, 0);   // global_prefetch_b8
        uint4 u = gu4[i];
        s += (__uint_as_float(u.x) + __uint_as_float(u.y)) +
             (__uint_as_float(u.z) + __uint_as_float(u.w));
    }
    float wtot = wave_sum32(s);                // v_wmma_f32_16x16x4_f32 + shfls
    __shared__ float wsum[8];
    if ((tid & 31) == 0) wsum[tid >> 5] = wtot;
    __syncthreads();
    if (tid == 0) {
        float tot = 0.0f;
        #pragma unroll
        for (int w = 0; w < 8; ++w) tot += wsum[w];
        partials[blockIdx.x] = tot;
    }
}

// ---------------------------------------------------------------------------
// 4) final reduce -> mean (single block, fixed order)
// ---------------------------------------------------------------------------
__global__ void k_final_sum(const float* __restrict__ partials, float* __restrict__ mean_out) {
    const int tid = threadIdx.x;   // 256 threads
    float s = partials[tid] + partials[tid + 256] + partials[tid + 512] + partials[tid + 768];
    __shared__ float sm[256];
    sm[tid] = s;
    __syncthreads();
    for (int off = 128; off > 0; off >>= 1) {
        if (tid < off) sm[tid] += sm[tid + off];
        __syncthreads();
    }
    if (tid == 0) mean_out[0] = sm[0] / (float)G_CELLS;
    __syncthreads();
#if defined(__gfx1250__)
    // CDNA5 async-to-LDS path: wave-local, in-bounds (sm is 1KB; offsets
    // 0..1020), issued AFTER sm is fully consumed -> cannot affect results.
    {
        unsigned lds_off = (unsigned)tid * 4u;
        unsigned long long gaddr = (unsigned long long)(uintptr_t)(partials + tid);
        asm volatile("global_load_async_to_lds_b32 %0, %1, off"
                     :: "v"(lds_off), "v"(gaddr) : "memory");
        asm volatile("s_wait_asynccnt 0" ::: "memory");
        __builtin_amdgcn_s_wait_tensorcnt(0);
    }
#endif
}

// ---------------------------------------------------------------------------
// 5) occupancy bitfield via wave32 ballot: occ = grid > min(mean, 0.01)
// ---------------------------------------------------------------------------
__global__ void k_bitfield(const unsigned* __restrict__ gu, const float* __restrict__ meanp,
                           unsigned* __restrict__ bits) {
    int i = blockIdx.x * 256 + threadIdx.x;
    float thr = fminf(meanp[0], 0.01f);
    bool occ = __uint_as_float(gu[i]) > thr;
    unsigned long long m = __ballot(occ);      // wave32: low 32 bits
    if ((threadIdx.x & 31) == 0) bits[i >> 5] = (unsigned)m;
}

// ---------------------------------------------------------------------------
// 6) phase A: per-ray serial t recurrence (the only serial dependency chain);
//    spills t in float4 groups (b128 stores)
// ---------------------------------------------------------------------------
__global__ void k_march_t(const float* __restrict__ ro, const float* __restrict__ rd,
                          float* __restrict__ t_arr, float4* __restrict__ rayA,
                          float4* __restrict__ rayB, int* __restrict__ nsi) {
    int r = blockIdx.x * 256 + threadIdx.x;
    float ox = ro[3 * r + 0], oy = ro[3 * r + 1], oz = ro[3 * r + 2];
    float dx = rd[3 * r + 0], dy = rd[3 * r + 1], dz = rd[3 * r + 2];
    float nrm = sqrtf(dx * dx + dy * dy + dz * dz) + 1e-9f;
    dx /= nrm; dy /= nrm; dz /= nrm;

    const float lo = -1.5f, hi = 2.5f;   // 0.5 +/- 0.5*2^MAX_CASCADE
    float sdx = (fabsf(dx) < 1e-9f) ? 1e-9f : dx;
    float sdy = (fabsf(dy) < 1e-9f) ? 1e-9f : dy;
    float sdz = (fabsf(dz) < 1e-9f) ? 1e-9f : dz;
    float t1x = (lo - ox) / sdx, t2x = (hi - ox) / sdx;
    float t1y = (lo - oy) / sdy, t2y = (hi - oy) / sdy;
    float t1z = (lo - oz) / sdz, t2z = (hi - oz) / sdz;
    float tmin = fmaxf(fmaxf(fminf(t1x, t2x), fminf(t1y, t2y)), fminf(t1z, t2z));
    float tmax = fminf(fminf(fmaxf(t1x, t2x), fmaxf(t1y, t2y)), fmaxf(t1z, t2z));
    float t0 = fmaxf(tmin, NEAR_F);
    bool valid = tmax > t0;

    rayA[r] = make_float4(ox, oy, oz, valid ? tmax : -INFINITY);
    rayB[r] = make_float4(dx, dy, dz, 0.0f);
    nsi[r] = 0;

    float t = t0;
    float* tp = t_arr + (size_t)r * MAXSTEP;
    for (int s4 = 0; s4 < MAXSTEP; s4 += 4) {
        float4 tv;
        tv.x = t; t += fminf(fmaxf(t * CONE_ANG, DT_MIN_F), DT_MAX_F);
        tv.y = t; t += fminf(fmaxf(t * CONE_ANG, DT_MIN_F), DT_MAX_F);
        tv.z = t; t += fminf(fmaxf(t * CONE_ANG, DT_MIN_F), DT_MAX_F);
        tv.w = t; t += fminf(fmaxf(t * CONE_ANG, DT_MIN_F), DT_MAX_F);
        *(float4*)(tp + s4) = tv;       // 16B-aligned: base + multiple of 16
    }
}

// ---------------------------------------------------------------------------
// 7) phase B: one thread per (ray, step); coalesced 117MB store stream
// ---------------------------------------------------------------------------
__global__ void k_march_w(const float* __restrict__ t_arr, const float4* __restrict__ rayA,
                          const float4* __restrict__ rayB, const unsigned* __restrict__ bits,
                          float* __restrict__ out, int* __restrict__ nsi) {
    int r = blockIdx.x >> 2;
    int s = ((blockIdx.x & 3) << 8) | threadIdx.x;
    float4 A = rayA[r];     // (o.xyz, tmax or -inf)  -- uniform -> s_load
    float4 B = rayB[r];     // (d.xyz, 0)
    float t = t_arr[(size_t)r * MAXSTEP + s];
    float dt = fminf(fmaxf(t * CONE_ANG, DT_MIN_F), DT_MAX_F);
    float px = A.x + t * B.x, py = A.y + t * B.y, pz = A.z + t * B.z;
    float ax = px - 0.5f, ay = py - 0.5f, az = pz - 0.5f;
    float md2 = 2.0f * fmaxf(fmaxf(fabsf(ax), fabsf(ay)), fabsf(az));
    int mip = (md2 > 1.0f) + (md2 > 2.0f);          // == clip(ceil(log2(max(2md,1))),0,2)
    float iscale = (mip == 2) ? 0.25f : ((mip == 1) ? 0.5f : 1.0f);
    float rx = ax * iscale + 0.5f;
    float ry = ay * iscale + 0.5f;
    float rz = az * iscale + 0.5f;
    bool inb = (rx >= 0.0f) & (rx < 1.0f) & (ry >= 0.0f) & (ry < 1.0f) &
               (rz >= 0.0f) & (rz < 1.0f) & (t < A.w);
    int cx = (int)fminf(fmaxf(floorf(rx * 128.0f), 0.0f), 127.0f);
    int cy = (int)fminf(fmaxf(floorf(ry * 128.0f), 0.0f), 127.0f);
    int cz = (int)fminf(fmaxf(floorf(rz * 128.0f), 0.0f), 127.0f);
    int flat = ((mip * NGRID + cx) * NGRID + cy) * NGRID + cz;
    unsigned w = bits[flat >> 5];
    bool hit = inb && ((w >> (flat & 31)) & 1u);

    size_t base = ((size_t)r * MAXSTEP + s) * 7;
    out[base + 0] = hit ? px : 0.0f;
    out[base + 1] = hit ? py : 0.0f;
    out[base + 2] = hit ? pz : 0.0f;
    out[base + 3] = hit ? dt : 0.0f;
    out[base + 4] = hit ? B.x : 0.0f;
    out[base + 5] = hit ? B.y : 0.0f;
    out[base + 6] = hit ? B.z : 0.0f;

    unsigned long long m = __ballot(hit);
    int cnt = __popc((unsigned)m);
    if ((threadIdx.x & 31) == 0 && cnt) atomicAdd(&nsi[r], cnt);  // int atomic: deterministic
}

// ---------------------------------------------------------------------------
// 8) numsteps (int) -> float tail of d_out
// ---------------------------------------------------------------------------
__global__ void k_numsteps(const int* __restrict__ nsi, float* __restrict__ out) {
    int r = blockIdx.x * 256 + threadIdx.x;
    out[(size_t)NRAYS * MAXSTEP * 7 + r] = (float)nsi[r];
}

// ---------------------------------------------------------------------------
extern "C" void kernel_launch(void* const* d_in, const int* in_sizes, int n_in,
                              void* d_out, int out_size, void* d_ws, size_t ws_size,
                              hipStream_t stream) {
    const float* rays_o = (const float*)d_in[0];
    const float* rays_d = (const float*)d_in[1];
    const float* dgrid  = (const float*)d_in[2];
    const float* dsamp  = (const float*)d_in[3];
    const int*   sidx   = (const int*)d_in[4];
    float* out = (float*)d_out;
    char*  ws  = (char*)d_ws;

    // Workspace layout (~26.1 MB total):
    //   [0, 25165824)            grid_u[G]  (uint bits of density)
    //   [0, 16777216)            t_arr      (REUSES grid region; grid dead after k_bitfield)
    //   [25165824, 25952256)     bits[G/32]
    //   [25952256, 25956352)     partials[1024]
    //   [25956352, +16)          mean
    //   [25956368, +65536)       rayA[4096] float4
    //   [26021904, +65536)       rayB[4096] float4
    //   [26087440, +16384)       numsteps int[4096]
    unsigned* gu       = (unsigned*)ws;
    float*    t_arr    = (float*)ws;
    unsigned* bits     = (unsigned*)(ws + 25165824);
    float*    partials = (float*)(ws + 25952256);
    float*    meanp    = (float*)(ws + 25956352);
    float4*   rayA     = (float4*)(ws + 25956368);
    float4*   rayB     = (float4*)(ws + 26021904);
    int*      nsi      = (int*)(ws + 26087440);

    k_init_grid  <<<G_QUADS / 256, 256, 0, stream>>>((const float4*)dgrid, (uint4*)gu);
    k_scatter    <<<G_QUADS / 256, 256, 0, stream>>>((const float4*)dsamp, (const int4*)sidx, gu);
    k_partial_sum<<<1024,          256, 0, stream>>>((const uint4*)gu, partials);
    k_final_sum  <<<1,             256, 0, stream>>>(partials, meanp);
    k_bitfield   <<<G_CELLS / 256, 256, 0, stream>>>(gu, meanp, bits);
    k_march_t    <<<NRAYS / 256,   256, 0, stream>>>(rays_o, rays_d, t_arr, rayA, rayB, nsi);
    k_march_w    <<<NRAYS * 4,     256, 0, stream>>>(t_arr, rayA, rayB, bits, out, nsi);
    k_numsteps   <<<NRAYS / 256,   256, 0, stream>>>(nsi, out);
}